// Mask2FormerHungarianMatcher_18597208391812
// MI455X (gfx1250) — compile-verified
//
#include <hip/hip_runtime.h>
#include <hip/hip_bf16.h>

#define BN   4      // batches
#define QN   200    // queries
#define TN   100    // targets
#define QP   208    // padded queries (13 * 16)
#define TP_  112    // padded targets (7 * 16)
#define HH   256
#define WW   256
#define PN   12544  // points == K reduction dim (divisible by 8*16)
#define LN   134    // classes

typedef float v2f __attribute__((ext_vector_type(2)));
typedef float v8f __attribute__((ext_vector_type(8)));

__device__ __forceinline__ float fast_sigmoid(float x) {
    // v_rcp_f32-based; avoids the IEEE divide expansion
    return __builtin_amdgcn_rcpf(1.0f + __expf(-x));
}

// ---------------------------------------------------------------------------
// 1) Bilinear point sampling: pp[b][q][p], sg=sigmoid(pp), tp[b][t][p]
//    (padded rows written deterministically; GEMM loads need no predication)
// ---------------------------------------------------------------------------
__global__ __launch_bounds__(256)
void sample_kernel(const float* __restrict__ mq, const float* __restrict__ ml,
                   const float* __restrict__ pc,
                   float* __restrict__ pp, float* __restrict__ sg,
                   float* __restrict__ tp)
{
    const int n = blockIdx.x;         // 0..QP+TP_-1
    const int b = blockIdx.y;
    const bool isPred = (n < QP);
    const int row = isPred ? n : (n - QP);
    const bool valid = isPred ? (row < QN) : (row < TN);
    float* outBase = isPred ? (pp + ((size_t)(b * QP + row)) * PN)
                            : (tp + ((size_t)(b * TP_ + row)) * PN);
    float* sgBase  = sg + ((size_t)(b * QP + (isPred ? row : 0))) * PN;
    const float* mask = isPred ? (mq + ((size_t)(b * QN + row)) * (HH * WW))
                               : (ml + ((size_t)(b * TN + row)) * (HH * WW));
    const float* pts = pc + (size_t)b * PN * 2;

    for (int p = threadIdx.x; p < PN; p += 256) {
        float val = 0.0f;
        if (valid) {
            float x = pts[2 * p + 0];
            float y = pts[2 * p + 1];
            float gx = x * (float)WW - 0.5f;
            float gy = y * (float)HH - 0.5f;
            float x0f = floorf(gx), y0f = floorf(gy);
            float wx = gx - x0f, wy = gy - y0f;
            int x0 = (int)x0f, y0 = (int)y0f;
            auto corner = [&](int xc, int yc, float w) -> float {
                if (xc < 0 || xc > WW - 1 || yc < 0 || yc > HH - 1) return 0.0f;
                return mask[yc * WW + xc] * w;
            };
            val = corner(x0,     y0,     (1.f - wx) * (1.f - wy))
                + corner(x0 + 1, y0,     wx         * (1.f - wy))
                + corner(x0,     y0 + 1, (1.f - wx) * wy)
                + corner(x0 + 1, y0 + 1, wx         * wy);
        }
        outBase[p] = val;
        if (isPred) sgBase[p] = fast_sigmoid(val);
    }
}

// ---------------------------------------------------------------------------
// 2) Row sums: ssp=Σ softplus(pp), ssg=Σ sg, stp=Σ tp
// ---------------------------------------------------------------------------
__global__ __launch_bounds__(256)
void sums_kernel(const float* __restrict__ pp, const float* __restrict__ sg,
                 const float* __restrict__ tp,
                 float* __restrict__ ssp, float* __restrict__ ssg,
                 float* __restrict__ stp)
{
    __shared__ float l1[256], l2[256];
    const int r = blockIdx.x;         // 0..BN*QN-1 pred rows, then BN*TN tgt
    const int tid = threadIdx.x;
    float a1 = 0.f, a2 = 0.f;
    if (r < BN * QN) {
        int b = r / QN, q = r % QN;
        const size_t ro = ((size_t)(b * QP + q)) * PN;
        const float* rp = pp + ro;
        const float* rs = sg + ro;
        for (int p = tid; p < PN; p += 256) {
            float x = rp[p];
            a1 += fmaxf(x, 0.f) + log1pf(__expf(-fabsf(x)));   // softplus
            a2 += rs[p];                                        // sigmoid (precomputed)
        }
    } else {
        int rr = r - BN * QN;
        int b = rr / TN, t = rr % TN;
        const float* rowp = tp + ((size_t)(b * TP_ + t)) * PN;
        for (int p = tid; p < PN; p += 256) a1 += rowp[p];
    }
    l1[tid] = a1; l2[tid] = a2;
    __syncthreads();
    for (int s = 128; s > 0; s >>= 1) {
        if (tid < s) { l1[tid] += l1[tid + s]; l2[tid] += l2[tid + s]; }
        __syncthreads();
    }
    if (tid == 0) {
        if (r < BN * QN) { ssp[r] = l1[0]; ssg[r] = l2[0]; }
        else             { stp[r - BN * QN] = l1[0]; }
    }
}

// ---------------------------------------------------------------------------
// 3) Class cost: cc[b][q][t] = -softmax(cls[b][q])[labels[b][t]]
// ---------------------------------------------------------------------------
__global__ __launch_bounds__(128)
void class_cost_kernel(const float* __restrict__ cls, const int* __restrict__ labels,
                       float* __restrict__ cc)
{
    __shared__ float red[128];
    const int q = blockIdx.x, b = blockIdx.y;
    const int tid = threadIdx.x;
    const float* lg = cls + ((size_t)(b * QN + q)) * LN;

    float m = -3.0e38f;
    for (int i = tid; i < LN; i += 128) m = fmaxf(m, lg[i]);
    red[tid] = m;
    __syncthreads();
    for (int s = 64; s > 0; s >>= 1) {
        if (tid < s) red[tid] = fmaxf(red[tid], red[tid + s]);
        __syncthreads();
    }
    float mx = red[0];
    __syncthreads();
    float su = 0.f;
    for (int i = tid; i < LN; i += 128) su += __expf(lg[i] - mx);
    red[tid] = su;
    __syncthreads();
    for (int s = 64; s > 0; s >>= 1) {
        if (tid < s) red[tid] += red[tid + s];
        __syncthreads();
    }
    float inv = 1.f / red[0];
    if (tid < TN) {
        int lbl = labels[b * TN + tid];
        cc[((size_t)(b * QN + q)) * TN + tid] = -__expf(lg[lbl] - mx) * inv;
    }
}

// ---------------------------------------------------------------------------
// 4) Dual f32 WMMA GEMM over K=PN + fused cost epilogue.
//    G1 = pp . tp^T ; G2 = sg . tp^T ; tile 16x16, 8 waves split K.
//    Inner loop: 12 global_load_b64 + 8 v_wmma per 16-k chunk (no divides).
// ---------------------------------------------------------------------------
__global__ __launch_bounds__(256)
void cost_gemm_kernel(const float* __restrict__ pp, const float* __restrict__ sg,
                      const float* __restrict__ tp,
                      const float* __restrict__ ssp, const float* __restrict__ ssg,
                      const float* __restrict__ stp, const float* __restrict__ cc,
                      float* __restrict__ Cout)
{
    const int tile_m = blockIdx.x;   // 0..12
    const int tile_n = blockIdx.y;   // 0..6
    const int b      = blockIdx.z;
    const int tid  = threadIdx.x;
    const int wave = tid >> 5;
    const int lane = tid & 31;

    const int KW = PN / 8;           // 1568 per wave, multiple of 16
    const int k0 = wave * KW;

    // A: 16x4 f32 layout: lanes 0-15 hold M=lane, K={0,1}; lanes 16-31 M=lane-16, K={2,3}
    const size_t aoff = ((size_t)(b * QP  + tile_m * 16 + (lane & 15))) * PN + ((lane >> 4) << 1);
    const float* Arow = pp + aoff;
    const float* Srow = sg + aoff;
    // B: 4x16 f32 (tp row n supplies column n of B): mirrored K split across half-waves
    const float* Brow = tp + ((size_t)(b * TP_ + tile_n * 16 + (lane & 15))) * PN + ((lane >> 4) << 1);

    v8f acc1 = {};   // pp . tp^T
    v8f acc2 = {};   // sigmoid(pp) . tp^T

    for (int k = k0; k < k0 + KW; k += 16) {
        // prefetch one chunk ahead (global_prefetch_b8)
        __builtin_prefetch(Arow + k + 16, 0, 1);
        __builtin_prefetch(Srow + k + 16, 0, 1);
        __builtin_prefetch(Brow + k + 16, 0, 1);

        v2f a0 = *(const v2f*)(Arow + k +  0);
        v2f a1 = *(const v2f*)(Arow + k +  4);
        v2f a2 = *(const v2f*)(Arow + k +  8);
        v2f a3 = *(const v2f*)(Arow + k + 12);
        v2f s0 = *(const v2f*)(Srow + k +  0);
        v2f s1 = *(const v2f*)(Srow + k +  4);
        v2f s2 = *(const v2f*)(Srow + k +  8);
        v2f s3 = *(const v2f*)(Srow + k + 12);
        v2f b0 = *(const v2f*)(Brow + k +  0);
        v2f b1 = *(const v2f*)(Brow + k +  4);
        v2f b2 = *(const v2f*)(Brow + k +  8);
        v2f b3 = *(const v2f*)(Brow + k + 12);

        acc1 = __builtin_amdgcn_wmma_f32_16x16x4_f32(false, a0, false, b0, (short)0, acc1, false, false);
        acc2 = __builtin_amdgcn_wmma_f32_16x16x4_f32(false, s0, false, b0, (short)0, acc2, false, false);
        acc1 = __builtin_amdgcn_wmma_f32_16x16x4_f32(false, a1, false, b1, (short)0, acc1, false, false);
        acc2 = __builtin_amdgcn_wmma_f32_16x16x4_f32(false, s1, false, b1, (short)0, acc2, false, false);
        acc1 = __builtin_amdgcn_wmma_f32_16x16x4_f32(false, a2, false, b2, (short)0, acc1, false, false);
        acc2 = __builtin_amdgcn_wmma_f32_16x16x4_f32(false, s2, false, b2, (short)0, acc2, false, false);
        acc1 = __builtin_amdgcn_wmma_f32_16x16x4_f32(false, a3, false, b3, (short)0, acc1, false, false);
        acc2 = __builtin_amdgcn_wmma_f32_16x16x4_f32(false, s3, false, b3, (short)0, acc2, false, false);
    }

    // Cross-wave reduction of both accumulators
    __shared__ float red[8 * 512];
#pragma unroll
    for (int v = 0; v < 8; v++) {
        red[wave * 512 +       v * 32 + lane] = acc1[v];
        red[wave * 512 + 256 + v * 32 + lane] = acc2[v];
    }
    __syncthreads();

    // Each thread owns one (m,n): element idx = v*32+lane (C layout: M=v+8*(lane>=16), N=lane&15)
    {
        const int v  = tid >> 5;
        const int ln = tid & 31;
        float g1 = 0.f, g2 = 0.f;
#pragma unroll
        for (int w = 0; w < 8; w++) {
            g1 += red[w * 512 + tid];
            g2 += red[w * 512 + 256 + tid];
        }
        const int m = tile_m * 16 + v + ((ln >> 4) << 3);
        const int n = tile_n * 16 + (ln & 15);
        if (m < QN && n < TN) {
            float sp = ssp[b * QN + m];
            float sgs = ssg[b * QN + m];
            float st = stp[b * TN + n];
            float cmask  = (sp - g1) * (1.0f / (float)PN);
            float cdice  = 1.0f - (2.0f * g2 + 1.0f) / (sgs + st + 1.0f);
            float cclass = cc[((size_t)(b * QN + m)) * TN + n];
            float cost = 5.0f * cmask + 2.0f * cclass + 5.0f * cdice;
            cost = fminf(fmaxf(cost, -1.0e10f), 1.0e10f);
            Cout[((size_t)(b * QN + m)) * TN + n] = cost;
        }
    }
}

// ---------------------------------------------------------------------------
// 5) Exact Hungarian (e-maxx) on transposed 100x200 matrix; one WG per batch.
// ---------------------------------------------------------------------------
__global__ __launch_bounds__(256)
void hungarian_kernel(const float* __restrict__ C,
                      float* __restrict__ rows_out, float* __restrict__ cols_out)
{
    const int b = blockIdx.x;
    const int tid = threadIdx.x;
    const float* base = C + (size_t)b * QN * TN;   // C[q][t]
    const int n = TN;   // rows of transposed matrix (targets)
    const int m = QN;   // cols (queries)

    __shared__ float u[TN + 1];
    __shared__ float v[QN + 1];
    __shared__ float minv[QN + 1];
    __shared__ int   p[QN + 1];
    __shared__ int   way[QN + 1];
    __shared__ unsigned char used[QN + 1];
    __shared__ int   s_j0, s_i0, s_j1, s_done;
    __shared__ float s_delta;
    __shared__ float rv[256];
    __shared__ int   ri[256];

    const float INF = 3.0e38f;

    for (int j = tid; j <= m; j += 256) { v[j] = 0.f; p[j] = 0; way[j] = 0; }
    for (int i = tid; i <= n; i += 256) u[i] = 0.f;
    __syncthreads();

    for (int i = 1; i <= n; i++) {
        if (tid == 0) { p[0] = i; s_j0 = 0; }
        for (int j = tid; j <= m; j += 256) { minv[j] = INF; used[j] = 0; }
        __syncthreads();

        while (true) {
            if (tid == 0) { used[s_j0] = 1; s_i0 = p[s_j0]; }
            __syncthreads();
            const int i0 = s_i0, j0 = s_j0;

            // transposed a[i0][j] = C[q=j-1][t=i0-1]
            for (int j = tid + 1; j <= m; j += 256) {
                if (!used[j]) {
                    float cur = base[(j - 1) * TN + (i0 - 1)] - u[i0] - v[j];
                    if (cur < minv[j]) { minv[j] = cur; way[j] = j0; }
                }
            }
            __syncthreads();

            // argmin over free columns
            float bv = INF; int bj = 1;
            for (int j = tid + 1; j <= m; j += 256)
                if (!used[j] && minv[j] < bv) { bv = minv[j]; bj = j; }
            rv[tid] = bv; ri[tid] = bj;
            __syncthreads();
            for (int s = 128; s > 0; s >>= 1) {
                if (tid < s) {
                    if (rv[tid + s] < rv[tid]) { rv[tid] = rv[tid + s]; ri[tid] = ri[tid + s]; }
                }
                __syncthreads();
            }
            if (tid == 0) { s_j1 = ri[0]; s_delta = rv[0]; }
            __syncthreads();
            const int j1 = s_j1; const float delta = s_delta;

            for (int j = tid; j <= m; j += 256) {
                if (used[j]) { u[p[j]] += delta; v[j] -= delta; }
                else         { minv[j] -= delta; }
            }
            __syncthreads();
            if (tid == 0) { s_j0 = j1; s_done = (p[j1] == 0); }
            __syncthreads();
            if (s_done) break;
        }

        if (tid == 0) {
            int j0 = s_j0;
            while (j0) { int j1 = way[j0]; p[j0] = p[j1]; j0 = j1; }
        }
        __syncthreads();
    }

    // Un-transpose: pairs (q=j-1, t=p[j]-1); ascending j == ascending q (stable)
    if (tid == 0) {
        int idx = 0;
        for (int j = 1; j <= m; j++) {
            if (p[j] > 0) {
                rows_out[b * TN + idx] = (float)(j - 1);
                cols_out[b * TN + idx] = (float)(p[j] - 1);
                idx++;
            }
        }
    }
}

// ---------------------------------------------------------------------------
// Launch
// ---------------------------------------------------------------------------
extern "C" void kernel_launch(void* const* d_in, const int* in_sizes, int n_in,
                              void* d_out, int out_size, void* d_ws, size_t ws_size,
                              hipStream_t stream)
{
    (void)in_sizes; (void)n_in; (void)out_size; (void)ws_size;

    const float* mq = (const float*)d_in[0];   // (B,Q,H,W)
    const float* cq = (const float*)d_in[1];   // (B,Q,L)
    const float* ml = (const float*)d_in[2];   // (B,T,H,W)
    const int*   cl = (const int*)  d_in[3];   // (B,T)
    const float* pc = (const float*)d_in[4];   // (B,P,2)

    float* out  = (float*)d_out;
    float* C    = out;                          // B*Q*T = 80000
    float* rows = out + (size_t)BN * QN * TN;   // B*T   = 400
    float* cols = rows + (size_t)BN * TN;       // B*T   = 400

    float* ws = (float*)d_ws;
    size_t o = 0;
    float* pp  = ws + o; o += (size_t)BN * QP  * PN;
    float* sg  = ws + o; o += (size_t)BN * QP  * PN;
    float* tp  = ws + o; o += (size_t)BN * TP_ * PN;
    float* ssp = ws + o; o += (size_t)BN * QN;
    float* ssg = ws + o; o += (size_t)BN * QN;
    float* stp = ws + o; o += (size_t)BN * TN;
    float* cc  = ws + o; o += (size_t)BN * QN * TN;

    sample_kernel<<<dim3(QP + TP_, BN), 256, 0, stream>>>(mq, ml, pc, pp, sg, tp);
    sums_kernel<<<dim3(BN * QN + BN * TN), 256, 0, stream>>>(pp, sg, tp, ssp, ssg, stp);
    class_cost_kernel<<<dim3(QN, BN), 128, 0, stream>>>(cq, cl, cc);
    cost_gemm_kernel<<<dim3(QP / 16, TP_ / 16, BN), 256, 0, stream>>>(pp, sg, tp, ssp, ssg, stp, cc, C);
    hungarian_kernel<<<dim3(BN), 256, 0, stream>>>(C, rows, cols);
}